// VectorQuantizer_58987080843732
// MI455X (gfx1250) — compile-verified
//
#include <hip/hip_runtime.h>
#include <hip/hip_bf16.h>

typedef float v2f __attribute__((ext_vector_type(2)));
typedef float v8f __attribute__((ext_vector_type(8)));

#define NUM_EMB   2048
#define EMB_DIM   128
#define ROWS_PER_BLOCK 64
#define LDS_STRIDE 132          // 128 + 4 pad: rows 16B aligned, conflict-free b64 reads
#define N_ROWS    65536         // 64*32*32
#define N_ELEMS   8388608       // N_ROWS * EMB_DIM
#define MAIN_BLOCKS 1024        // N_ROWS / ROWS_PER_BLOCK

// ---------------------------------------------------------------------------
// Kernel 1: per-code squared norms  ||e_k||^2
// ---------------------------------------------------------------------------
__global__ __launch_bounds__(256) void vq_enorm_kernel(const float* __restrict__ emb,
                                                       float* __restrict__ enorm) {
    int k = blockIdx.x * 256 + threadIdx.x;      // 0..2047
    const float4* p = (const float4*)(emb + (size_t)k * EMB_DIM);
    float s = 0.f;
#pragma unroll 8
    for (int i = 0; i < EMB_DIM / 4; ++i) {
        float4 v = p[i];
        s = fmaf(v.x, v.x, s);
        s = fmaf(v.y, v.y, s);
        s = fmaf(v.z, v.z, s);
        s = fmaf(v.w, v.w, s);
    }
    enorm[k] = s;
}

// ---------------------------------------------------------------------------
// Kernel 2: fused WMMA GEMM (x . e^T) + argmin + gather + loss partials
//   block = 64 rows of z, 8 waves, each wave covers 16 tiles of 16 codes
// ---------------------------------------------------------------------------
__global__ __launch_bounds__(256) void vq_main_kernel(const float* __restrict__ z,
                                                      const float* __restrict__ emb,
                                                      const float* __restrict__ enorm,
                                                      float* __restrict__ qout,
                                                      float* __restrict__ iout,
                                                      float* __restrict__ partial) {
    __shared__ float zlds[ROWS_PER_BLOCK * LDS_STRIDE];   // 33792 B
    __shared__ float minv_s[ROWS_PER_BLOCK * 8];
    __shared__ int   mini_s[ROWS_PER_BLOCK * 8];
    __shared__ int   fidx_s[ROWS_PER_BLOCK];
    __shared__ float red_s[256];

    const int tid  = threadIdx.x;
    const int lane = tid & 31;
    const int wave = tid >> 5;
    const size_t rowbase = (size_t)blockIdx.x * ROWS_PER_BLOCK;

    // ---- stage 64x128 z tile into padded LDS (coalesced float4) ----
    const float4* zg = (const float4*)(z + rowbase * EMB_DIM);
#pragma unroll
    for (int i = 0; i < 8; ++i) {
        int idx = tid + i * 256;                 // 0..2047 float4s
        int r = idx >> 5;                        // 32 float4 per row
        int c = (idx & 31) << 2;
        float4 v = zg[idx];
        *(float4*)&zlds[r * LDS_STRIDE + c] = v;
    }

    const int bn   = lane & 15;                  // column within 16-wide tile
    const int koff = (lane >> 4) << 1;           // 0 (lanes 0-15) or 2 (lanes 16-31)

    // ---- prefetch all 16 per-tile ||e||^2 values for this wave's columns ----
    float en_t[16];
#pragma unroll
    for (int t = 0; t < 16; ++t)
        en_t[t] = enorm[((wave + (t << 3)) << 4) + bn];

    __syncthreads();

    // ---- running argmin state: 4 M-tiles x 8 accumulator rows per lane ----
    float bestv[4][8];
    int   besti[4][8];
#pragma unroll
    for (int mt = 0; mt < 4; ++mt)
#pragma unroll
        for (int v = 0; v < 8; ++v) { bestv[mt][v] = 3.402823466e38f; besti[mt][v] = 0; }

    const float* ap = &zlds[bn * LDS_STRIDE + koff];

    for (int t = 0; t < 16; ++t) {
        const int j = (wave + (t << 3)) << 4;    // tile base code index
        const v2f* bp = (const v2f*)(emb + (size_t)(j + bn) * EMB_DIM + koff);

        v8f acc[4];
#pragma unroll
        for (int mt = 0; mt < 4; ++mt) acc[mt] = (v8f)0.f;

        // K = 128 = 32 steps of K=4  ->  128 v_wmma_f32_16x16x4_f32 per tile
#pragma unroll 8
        for (int s = 0; s < 32; ++s) {
            v2f b = bp[s * 2];                   // emb[j+bn][4s+koff .. +1]
#pragma unroll
            for (int mt = 0; mt < 4; ++mt) {
                v2f a = *(const v2f*)(ap + mt * (16 * LDS_STRIDE) + s * 4);
                acc[mt] = __builtin_amdgcn_wmma_f32_16x16x4_f32(
                    false, a, false, b, (short)0, acc[mt], false, false);
            }
        }

        // epilogue: dist' = ||e||^2 - 2 x.e ; branch-free running argmin
        const float en  = en_t[t];
        const int   col = j + bn;
#pragma unroll
        for (int mt = 0; mt < 4; ++mt)
#pragma unroll
            for (int v = 0; v < 8; ++v) {
                float val = fmaf(-2.0f, acc[mt][v], en);
                float bv  = bestv[mt][v];
                int   bi  = besti[mt][v];
                bool better = (val < bv) | ((val == bv) & (col < bi));
                bestv[mt][v] = better ? val : bv;
                besti[mt][v] = better ? col : bi;
            }
    }

    // ---- cross-lane argmin within each 16-lane half (xor masks 1,2,4,8) ----
#pragma unroll
    for (int mt = 0; mt < 4; ++mt)
#pragma unroll
        for (int v = 0; v < 8; ++v) {
            float bv = bestv[mt][v];
            int   bi = besti[mt][v];
#pragma unroll
            for (int m = 1; m < 16; m <<= 1) {
                float ov = __shfl_xor(bv, m, 32);
                int   oi = __shfl_xor(bi, m, 32);
                bool better = (ov < bv) | ((ov == bv) & (oi < bi));
                bv = better ? ov : bv;
                bi = better ? oi : bi;
            }
            if ((lane & 15) == 0) {
                // lanes 0-15 hold rows mt*16+v, lanes 16-31 hold rows mt*16+8+v
                int row = mt * 16 + ((lane >> 4) << 3) + v;
                minv_s[row * 8 + wave] = bv;
                mini_s[row * 8 + wave] = bi;
            }
        }
    __syncthreads();

    // ---- cross-wave argmin, emit indices ----
    if (tid < ROWS_PER_BLOCK) {
        float bv = minv_s[tid * 8];
        int   bi = mini_s[tid * 8];
#pragma unroll
        for (int w = 1; w < 8; ++w) {
            float ov = minv_s[tid * 8 + w];
            int   oi = mini_s[tid * 8 + w];
            bool better = (ov < bv) | ((ov == bv) & (oi < bi));
            bv = better ? ov : bv;
            bi = better ? oi : bi;
        }
        fidx_s[tid] = bi;
        iout[rowbase + tid] = (float)bi;
    }
    __syncthreads();

    // ---- gather quantized = emb[idx], accumulate (q - z)^2 ----
    float lsum = 0.f;
#pragma unroll 4
    for (int i = 0; i < 32; ++i) {
        int e = tid + i * 256;                   // 0..8191
        int r = e >> 7;
        int c = e & 127;
        float q = emb[(size_t)fidx_s[r] * EMB_DIM + c];
        qout[rowbase * EMB_DIM + e] = q;
        float d = q - zlds[r * LDS_STRIDE + c];
        lsum = fmaf(d, d, lsum);
    }
    red_s[tid] = lsum;
    __syncthreads();
#pragma unroll
    for (int st = 128; st > 0; st >>= 1) {
        if (tid < st) red_s[tid] += red_s[tid + st];
        __syncthreads();
    }
    if (tid == 0) partial[blockIdx.x] = red_s[0];
}

// ---------------------------------------------------------------------------
// Kernel 3: deterministic final loss reduction
//   loss = (1 + beta) * SSD / N_ELEMS,  beta = 0.25
// ---------------------------------------------------------------------------
__global__ __launch_bounds__(256) void vq_loss_kernel(const float* __restrict__ partial,
                                                      float* __restrict__ lout) {
    __shared__ float r[256];
    float s = 0.f;
    for (int i = threadIdx.x; i < MAIN_BLOCKS; i += 256) s += partial[i];
    r[threadIdx.x] = s;
    __syncthreads();
#pragma unroll
    for (int st = 128; st > 0; st >>= 1) {
        if (threadIdx.x < st) r[threadIdx.x] += r[threadIdx.x + st];
        __syncthreads();
    }
    if (threadIdx.x == 0) lout[0] = 1.25f * r[0] / (float)N_ELEMS;
}

// ---------------------------------------------------------------------------
extern "C" void kernel_launch(void* const* d_in, const int* in_sizes, int n_in,
                              void* d_out, int out_size, void* d_ws, size_t ws_size,
                              hipStream_t stream) {
    const float* z   = (const float*)d_in[0];   // [64,32,32,128]
    const float* emb = (const float*)d_in[1];   // [2048,128]

    float* out  = (float*)d_out;
    float* qout = out;                          // 8388608 floats (quantized_st)
    float* iout = out + N_ELEMS;                // 65536 floats (indices)
    float* lout = out + N_ELEMS + N_ROWS;       // 1 float (loss)

    float* enorm   = (float*)d_ws;              // 2048 floats
    float* partial = enorm + NUM_EMB;           // 1024 floats

    vq_enorm_kernel<<<NUM_EMB / 256, 256, 0, stream>>>(emb, enorm);
    vq_main_kernel<<<MAIN_BLOCKS, 256, 0, stream>>>(z, emb, enorm, qout, iout, partial);
    vq_loss_kernel<<<1, 256, 0, stream>>>(partial, lout);
}